// QFCModel_65481071407946
// MI455X (gfx1250) — compile-verified
//
#include <hip/hip_runtime.h>
#include <math.h>

// ---------------------------------------------------------------------------
// QFCModel on MI455X (gfx1250, wave32)
//
// out_f(sample) = s0^T Q_f s0 + b_f, then batch-norm.
//   s0 = kron_i [cos(a_i/2), sin(a_i/2)]  (16-vector, real)
//   Q_f (16x16, real symmetric) folded once per call from weights/fc_w.
// Main kernel: per wave, 16 samples; T = Q_f @ S via 4x V_WMMA_F32_16X16X4_F32
// (K chunks), then rowwise dot with S (8 FMA + shfl_xor 16).
// ---------------------------------------------------------------------------

#define NQ 4
#define NL 3
#define B_TOTAL 524288
#define PI_F 3.14159265358979f
#define BN_EPS 1e-5f

typedef __attribute__((ext_vector_type(2))) float v2f;
typedef __attribute__((ext_vector_type(8))) float v8f;

// ws layout (float offsets)
#define WS_MN    0
#define WS_MX    1
#define WS_SCALE 8      // 4 floats
#define WS_SHIFT 12     // 4 floats
#define WS_Q     256    // 4*16*16 = 1024 floats
#define WS_MM    1536   // 256 blocks * 2 floats
#define WS_STAT  2048   // 1024 blocks * 8 floats

#define MM_BLOCKS   256
#define MAIN_BLOCKS 1024
#define MAIN_ITERS  4   // MAIN_BLOCKS * 128 * MAIN_ITERS == B_TOTAL

// ---------------------------------------------------------------------------
// 1) Per-block min/max over x[:, 0:4]
// ---------------------------------------------------------------------------
__global__ __launch_bounds__(256) void k_minmax(const float* __restrict__ x,
                                                float* __restrict__ ws) {
  int g = blockIdx.x * 256 + threadIdx.x;
  float mn = INFINITY, mx = -INFINITY;
  for (int row = g; row < B_TOTAL; row += MM_BLOCKS * 256) {
    const float4 v = *(const float4*)(x + (size_t)row * 16);
    mn = fminf(mn, fminf(fminf(v.x, v.y), fminf(v.z, v.w)));
    mx = fmaxf(mx, fmaxf(fmaxf(v.x, v.y), fmaxf(v.z, v.w)));
  }
  __shared__ float smn[256], smx[256];
  smn[threadIdx.x] = mn; smx[threadIdx.x] = mx;
  __syncthreads();
  for (int s = 128; s > 0; s >>= 1) {
    if (threadIdx.x < s) {
      smn[threadIdx.x] = fminf(smn[threadIdx.x], smn[threadIdx.x + s]);
      smx[threadIdx.x] = fmaxf(smx[threadIdx.x], smx[threadIdx.x + s]);
    }
    __syncthreads();
  }
  if (threadIdx.x == 0) {
    ws[WS_MM + 2 * blockIdx.x]     = smn[0];
    ws[WS_MM + 2 * blockIdx.x + 1] = smx[0];
  }
}

// ---------------------------------------------------------------------------
// 2) Setup: finalize min/max; simulate layer unitary U (16 cols); build Q_f.
// ---------------------------------------------------------------------------
__global__ __launch_bounds__(256) void k_setup(const float* __restrict__ weights,
                                               const float* __restrict__ fc_w,
                                               float* __restrict__ ws) {
  const int tid = threadIdx.x;
  __shared__ float smn[256], smx[256];
  smn[tid] = ws[WS_MM + 2 * tid];
  smx[tid] = ws[WS_MM + 2 * tid + 1];
  __syncthreads();
  for (int s = 128; s > 0; s >>= 1) {
    if (tid < s) {
      smn[tid] = fminf(smn[tid], smn[tid + s]);
      smx[tid] = fmaxf(smx[tid], smx[tid + s]);
    }
    __syncthreads();
  }
  if (tid == 0) { ws[WS_MN] = smn[0]; ws[WS_MX] = smx[0]; }

  // --- phase 1: thread j simulates column j of the layer unitary ---
  __shared__ float Ure[16][16], Uim[16][16];
  if (tid < 16) {
    const int j = tid;
    for (int k = 0; k < 16; ++k) { Ure[k][j] = (k == j) ? 1.f : 0.f; Uim[k][j] = 0.f; }
    for (int layer = 0; layer < NL; ++layer) {
      for (int q = 0; q < NQ; ++q) {
        const int mask = 8 >> q;                 // qubit q = bit (3-q)
        const float thY = weights[(layer * NQ + q) * 2 + 0];
        const float c = cosf(0.5f * thY), s = sinf(0.5f * thY);
        for (int k = 0; k < 16; ++k) {
          if (!(k & mask)) {
            const int k1 = k | mask;
            float a0r = Ure[k][j],  a0i = Uim[k][j];
            float a1r = Ure[k1][j], a1i = Uim[k1][j];
            Ure[k][j]  = c * a0r - s * a1r;  Uim[k][j]  = c * a0i - s * a1i;
            Ure[k1][j] = s * a0r + c * a1r;  Uim[k1][j] = s * a0i + c * a1i;
          }
        }
        const float thZ = weights[(layer * NQ + q) * 2 + 1];
        const float cz = cosf(0.5f * thZ), sz = sinf(0.5f * thZ);
        for (int k = 0; k < 16; ++k) {
          const float sgn = (k & mask) ? sz : -sz;   // e^{±i thZ/2}
          const float ar = Ure[k][j], ai = Uim[k][j];
          Ure[k][j] = cz * ar - sgn * ai;
          Uim[k][j] = cz * ai + sgn * ar;
        }
      }
      // CNOT ring: (0,1),(1,2),(2,3),(3,0)
      const int cq[4] = {0, 1, 2, 3}, tq[4] = {1, 2, 3, 0};
      for (int e = 0; e < 4; ++e) {
        const int mc = 8 >> cq[e], mt = 8 >> tq[e];
        for (int k = 0; k < 16; ++k) {
          if ((k & mc) && !(k & mt)) {
            const int k1 = k | mt;
            float tr = Ure[k][j], ti = Uim[k][j];
            Ure[k][j] = Ure[k1][j]; Uim[k][j] = Uim[k1][j];
            Ure[k1][j] = tr;        Uim[k1][j] = ti;
          }
        }
      }
    }
  }
  __syncthreads();

  // --- phase 2: Q_f[j][jp] = sum_i fc_w[f,i] * sum_k sign_i(k) Re(U[k,j] U*[k,jp])
  const int j = tid & 15, jp = tid >> 4;
  float r[16];
  #pragma unroll
  for (int k = 0; k < 16; ++k)
    r[k] = Ure[k][j] * Ure[k][jp] + Uim[k][j] * Uim[k][jp];
  float zc[4];
  #pragma unroll
  for (int i = 0; i < 4; ++i) {
    const int mask = 8 >> i;
    float acc = 0.f;
    #pragma unroll
    for (int k = 0; k < 16; ++k) acc += (k & mask) ? -r[k] : r[k];
    zc[i] = acc;
  }
  #pragma unroll
  for (int f = 0; f < 4; ++f) {
    const float qv = zc[0] * fc_w[f * 4 + 0] + zc[1] * fc_w[f * 4 + 1] +
                     zc[2] * fc_w[f * 4 + 2] + zc[3] * fc_w[f * 4 + 3];
    ws[WS_Q + f * 256 + j * 16 + jp] = qv;
  }
}

// ---------------------------------------------------------------------------
// 3) Main: per wave, 16 samples; out_f = s0^T Q_f s0 + b_f via FP32 WMMA.
//    Also per-block BN partial sums (deterministic).
// ---------------------------------------------------------------------------
__global__ __launch_bounds__(256) void k_main(const float* __restrict__ x,
                                              const float* __restrict__ fc_b,
                                              float* ws,
                                              float* __restrict__ out) {
  const int lane = threadIdx.x & 31;
  const int wid  = threadIdx.x >> 5;
  const int m    = lane & 15;          // A-matrix row / sample-in-wave
  const int kh   = lane >> 4;          // K-half select (ISA A/B operand layout)

  const float mn  = ws[WS_MN];
  const float mx  = ws[WS_MX];
  const float inv = PI_F / (mx - mn + 1e-8f);   // theta/2 = pi*(x-mn)/range

  // Hoist A operands (Q_f chunks) into registers: A[f][t] = Q[f][m][4t+2kh .. +1]
  v2f A[4][4];
  #pragma unroll
  for (int f = 0; f < 4; ++f)
    #pragma unroll
    for (int t = 0; t < 4; ++t) {
      const float* qp = ws + WS_Q + f * 256 + m * 16 + 4 * t + 2 * kh;
      A[f][t].x = qp[0];
      A[f][t].y = qp[1];
    }
  float bb[4];
  #pragma unroll
  for (int f = 0; f < 4; ++f) bb[f] = fc_b[f];

  float ssum[4] = {0.f, 0.f, 0.f, 0.f};
  float ssq[4]  = {0.f, 0.f, 0.f, 0.f};

  for (int it = 0; it < MAIN_ITERS; ++it) {
    const int sample = ((it * MAIN_BLOCKS) + blockIdx.x) * 128 + wid * 16 + m;
    const float4 xv = *(const float4*)(x + (size_t)sample * 16);
    if (it + 1 < MAIN_ITERS) {
      const int ns = (((it + 1) * MAIN_BLOCKS) + blockIdx.x) * 128 + wid * 16 + m;
      __builtin_prefetch(x + (size_t)ns * 16, 0, 0);   // global_prefetch_b8
    }

    const float a0 = (xv.x - mn) * inv, a1 = (xv.y - mn) * inv;
    const float a2 = (xv.z - mn) * inv, a3 = (xv.w - mn) * inv;
    const float c0 = __cosf(a0), s0v = __sinf(a0);
    const float c1 = __cosf(a1), s1v = __sinf(a1);
    const float c2 = __cosf(a2), s2v = __sinf(a2);
    const float c3 = __cosf(a3), s3v = __sinf(a3);

    // s0[j] = pA[(b0,b1)] * pB[(b2,b3)], qubit0 = MSB of j
    float pA0 = c0 * c1, pA1 = c0 * s1v, pA2 = s0v * c1, pA3 = s0v * s1v;
    float pB0 = c2 * c3, pB1 = c2 * s3v, pB2 = s2v * c3, pB3 = s2v * s3v;
    float S[16];
    S[0]  = pA0 * pB0; S[1]  = pA0 * pB1; S[2]  = pA0 * pB2; S[3]  = pA0 * pB3;
    S[4]  = pA1 * pB0; S[5]  = pA1 * pB1; S[6]  = pA1 * pB2; S[7]  = pA1 * pB3;
    S[8]  = pA2 * pB0; S[9]  = pA2 * pB1; S[10] = pA2 * pB2; S[11] = pA2 * pB3;
    S[12] = pA3 * pB0; S[13] = pA3 * pB1; S[14] = pA3 * pB2; S[15] = pA3 * pB3;

    float outv[4];
    #pragma unroll
    for (int f = 0; f < 4; ++f) {
      v8f acc = {};
      #pragma unroll
      for (int t = 0; t < 4; ++t) {
        v2f Bv;                                  // B: lanes<16 K=4t,4t+1; lanes>=16 K=4t+2,4t+3
        Bv.x = kh ? S[4 * t + 2] : S[4 * t + 0];
        Bv.y = kh ? S[4 * t + 3] : S[4 * t + 1];
        acc = __builtin_amdgcn_wmma_f32_16x16x4_f32(
            /*neg_a=*/false, A[f][t], /*neg_b=*/false, Bv,
            /*c_mod=*/(short)0, acc, /*reuse_a=*/false, /*reuse_b=*/false);
      }
      // rowwise dot: sum_m S[m][n] * D[m][n]; lane holds rows m = r + 8*kh
      float partial = 0.f;
      #pragma unroll
      for (int r = 0; r < 8; ++r) {
        const float sv = kh ? S[r + 8] : S[r];
        partial = fmaf(acc[r], sv, partial);
      }
      const float tot = partial + __shfl_xor(partial, 16, 32);
      outv[f] = tot + bb[f];
    }

    if (lane < 16) {
      *(float4*)(out + (size_t)sample * 4) =
          make_float4(outv[0], outv[1], outv[2], outv[3]);
      #pragma unroll
      for (int f = 0; f < 4; ++f) {
        ssum[f] += outv[f];
        ssq[f]  += outv[f] * outv[f];
      }
    }
  }

  // Deterministic per-block BN partial sums
  __shared__ float sred[8][256];
  #pragma unroll
  for (int f = 0; f < 4; ++f) {
    sred[f][threadIdx.x]     = ssum[f];
    sred[4 + f][threadIdx.x] = ssq[f];
  }
  __syncthreads();
  if (threadIdx.x < 8) {
    float a = 0.f;
    for (int k = 0; k < 256; ++k) a += sred[threadIdx.x][k];
    ws[WS_STAT + blockIdx.x * 8 + threadIdx.x] = a;
  }
}

// ---------------------------------------------------------------------------
// 4) Reduce BN partials -> scale/shift (fixed order, deterministic)
// ---------------------------------------------------------------------------
__global__ __launch_bounds__(256) void k_finalize(const float* __restrict__ gamma,
                                                  const float* __restrict__ beta,
                                                  float* __restrict__ ws) {
  __shared__ float tot[8];
  const int tid = threadIdx.x;
  if (tid < 8) {
    float a = 0.f;
    for (int b = 0; b < MAIN_BLOCKS; ++b) a += ws[WS_STAT + b * 8 + tid];
    tot[tid] = a;
  }
  __syncthreads();
  if (tid < 4) {
    const float invB  = 1.0f / (float)B_TOTAL;
    const float mean  = tot[tid] * invB;
    const float var   = tot[4 + tid] * invB - mean * mean;
    const float scale = gamma[tid] * rsqrtf(var + BN_EPS);
    ws[WS_SCALE + tid] = scale;
    ws[WS_SHIFT + tid] = beta[tid] - mean * scale;
  }
}

// ---------------------------------------------------------------------------
// 5) Apply batch-norm in place (float4 per sample)
// ---------------------------------------------------------------------------
__global__ __launch_bounds__(256) void k_norm(float* __restrict__ out,
                                              const float* __restrict__ ws) {
  const int sample = blockIdx.x * 256 + threadIdx.x;
  const float sc0 = ws[WS_SCALE + 0], sc1 = ws[WS_SCALE + 1];
  const float sc2 = ws[WS_SCALE + 2], sc3 = ws[WS_SCALE + 3];
  const float sh0 = ws[WS_SHIFT + 0], sh1 = ws[WS_SHIFT + 1];
  const float sh2 = ws[WS_SHIFT + 2], sh3 = ws[WS_SHIFT + 3];
  float4 o = *(float4*)(out + (size_t)sample * 4);
  o.x = fmaf(o.x, sc0, sh0);
  o.y = fmaf(o.y, sc1, sh1);
  o.z = fmaf(o.z, sc2, sh2);
  o.w = fmaf(o.w, sc3, sh3);
  *(float4*)(out + (size_t)sample * 4) = o;
}

// ---------------------------------------------------------------------------
extern "C" void kernel_launch(void* const* d_in, const int* in_sizes, int n_in,
                              void* d_out, int out_size, void* d_ws, size_t ws_size,
                              hipStream_t stream) {
  const float* x       = (const float*)d_in[0];
  const float* weights = (const float*)d_in[1];
  const float* fc_w    = (const float*)d_in[2];
  const float* fc_b    = (const float*)d_in[3];
  const float* gamma   = (const float*)d_in[4];
  const float* beta    = (const float*)d_in[5];
  float* out = (float*)d_out;
  float* ws  = (float*)d_ws;

  k_minmax  <<<MM_BLOCKS, 256, 0, stream>>>(x, ws);
  k_setup   <<<1,          256, 0, stream>>>(weights, fc_w, ws);
  k_main    <<<MAIN_BLOCKS,256, 0, stream>>>(x, fc_b, ws, out);
  k_finalize<<<1,          256, 0, stream>>>(gamma, beta, ws);
  k_norm    <<<B_TOTAL/256,256, 0, stream>>>(out, ws);
}